// HermiteActivation_8409545966398
// MI455X (gfx1250) — compile-verified
//
#include <hip/hip_runtime.h>

typedef __attribute__((ext_vector_type(2))) float v2f;
typedef __attribute__((ext_vector_type(8))) float v8f;

constexpr int BATCH = 4096;
constexpr int FEAT  = 1024;
constexpr int OUTF  = 1024;
constexpr int W     = 7;          // Hermite degrees 0..6
constexpr int ROWS_PER_BLOCK = 16;

// One block (8 wave32) owns 16 batch rows:
//  Phase A: basis[16][8] = sum_f H_k(x/s) * exp(-min((x/s)^2,50)), K padded to 8
//  Phase B: out[16 x 1024] tile-stripe via V_WMMA_F32_16X16X4_F32 (2 WMMAs / 16x16 tile)
__global__ __launch_bounds__(256)
void hermite_wmma_fused(const float* __restrict__ x,
                        const float* __restrict__ coeffs,
                        const float* __restrict__ sigma,
                        float* __restrict__ out) {
    __shared__ float basis[ROWS_PER_BLOCK][8];

    const int tid     = threadIdx.x;
    const int wave    = tid >> 5;          // 0..7
    const int lane    = tid & 31;
    const int rowBase = blockIdx.x * ROWS_PER_BLOCK;

    const float s    = fminf(fmaxf(sigma[0], 0.1f), 5.0f);
    const float invs = 1.0f / s;

    // ---------------- Phase A: feature reduction (2 rows per wave) -------------
    for (int rr = 0; rr < 2; ++rr) {
        const int r   = wave + rr * 8;     // row within tile (0..15)
        const int row = rowBase + r;
        const float4* xrow = reinterpret_cast<const float4*>(x + (size_t)row * FEAT);

        float acc[W];
#pragma unroll
        for (int k = 0; k < W; ++k) acc[k] = 0.0f;

#pragma unroll
        for (int it = 0; it < FEAT / 128; ++it) {   // 8 iters, 128-bit coalesced loads
            const float4 v = xrow[it * 32 + lane];
            const float e[4] = {v.x, v.y, v.z, v.w};
#pragma unroll
            for (int j = 0; j < 4; ++j) {
                const float xs = e[j] * invs;
                const float t  = xs + xs;                       // 2*xs
                const float g  = __expf(-fminf(xs * xs, 50.0f));
                float h0 = 1.0f;
                float h1 = t;
                acc[0] += g;
                acc[1] += h1 * g;
#pragma unroll
                for (int n = 2; n <= 6; ++n) {
                    float h = t * h1 - (2.0f * (float)(n - 1)) * h0;
                    h = fminf(fmaxf(h, -100.0f), 100.0f);
                    acc[n] += h * g;
                    h0 = h1; h1 = h;
                }
            }
        }

        // wave32 tree reduction
#pragma unroll
        for (int k = 0; k < W; ++k) {
#pragma unroll
            for (int off = 16; off > 0; off >>= 1)
                acc[k] += __shfl_xor(acc[k], off, 32);
        }
        if (lane == 0) {
#pragma unroll
            for (int k = 0; k < W; ++k) basis[r][k] = acc[k];
            basis[r][7] = 0.0f;                               // K padding
        }
    }
    __syncthreads();

    // ---------------- Phase B: out = basis(16x8) x coeffs^T(8x1024) ------------
    const int half = lane >> 4;      // 0: lanes 0-15, 1: lanes 16-31
    const int l16  = lane & 15;

    // A fragment, 16x4 f32 layout: VGPR0/1 hold K{0,1} (lanes 0-15), K{2,3} (lanes 16-31)
    v2f a_lo, a_hi;
    a_lo.x = basis[l16][2 * half + 0];
    a_lo.y = basis[l16][2 * half + 1];
    a_hi.x = basis[l16][4 + 2 * half + 0];
    a_hi.y = basis[l16][4 + 2 * half + 1];

    // 64 column tiles total; 8 per wave
    for (int ct = wave * 8; ct < wave * 8 + 8; ++ct) {
        const int o = ct * 16 + l16;                          // output column

        // B fragment, 4x16 f32 (KxN): VGPR0/1 hold K{0,1}/K{2,3} mirrored vs A.
        // K indices per lane-half: lo = {2h, 2h+1}, hi = {4+2h, 4+2h+1}.
        // Only k=7 (hi.y, upper half) is padding: load a valid dummy (k=6),
        // zero it with a select -> no EXEC divergence, no predicated loads.
        const int k0 = 2 * half;             // 0 or 2
        const int k4 = 4 + 2 * half;         // 4 or 6 (always valid)
        const int ky = half ? 6 : 5;         // valid address stand-in for k4+1
        v2f b_lo, b_hi;
        b_lo.x = coeffs[o * W + k0];
        b_lo.y = coeffs[o * W + k0 + 1];
        b_hi.x = coeffs[o * W + k4];
        const float by = coeffs[o * W + ky];
        b_hi.y = half ? 0.0f : by;           // v_cndmask, branch-free K padding

        v8f c = {};
        c = __builtin_amdgcn_wmma_f32_16x16x4_f32(false, a_lo, false, b_lo,
                                                  (short)0, c, false, false);
        c = __builtin_amdgcn_wmma_f32_16x16x4_f32(false, a_hi, false, b_hi,
                                                  (short)0, c, false, false);

        // C/D layout: lane -> N, VGPR v -> M = v + 8*half
#pragma unroll
        for (int v = 0; v < 8; ++v) {
            const int m = v + 8 * half;
            out[(size_t)(rowBase + m) * OUTF + o] = c[v];
        }
    }
}

extern "C" void kernel_launch(void* const* d_in, const int* in_sizes, int n_in,
                              void* d_out, int out_size, void* d_ws, size_t ws_size,
                              hipStream_t stream) {
    const float* x      = (const float*)d_in[0];
    const float* coeffs = (const float*)d_in[1];
    const float* sigma  = (const float*)d_in[2];
    float* out          = (float*)d_out;

    dim3 grid(BATCH / ROWS_PER_BLOCK);   // 256 blocks x 8 waves
    dim3 block(256);
    hermite_wmma_fused<<<grid, block, 0, stream>>>(x, coeffs, sigma, out);
}